// EmotionalGraphNN_27814208209785
// MI455X (gfx1250) — compile-verified
//
#include <hip/hip_runtime.h>
#include <math.h>

// Problem constants (match reference)
#define NN      100000      // nodes
#define NE      1600000     // edges
#define F_IN    128
#define HID     128
#define HID2    64
#define N_CLS   16

typedef __attribute__((ext_vector_type(2))) float v2f;
typedef __attribute__((ext_vector_type(8))) float v8f;

// 16-byte int vector type matching the async-to-LDS builtin's pointee type
typedef int v4i_b128 __attribute__((vector_size(16)));
typedef v4i_b128 __attribute__((address_space(1)))* g_b128_p;   // global (AS1)
typedef v4i_b128 __attribute__((address_space(3)))* l_b128_p;   // LDS    (AS3)

static inline int cdiv(long long a, long long b) { return (int)((a + b - 1) / b); }

// ---------------- CDNA5 async global->LDS copy helpers ----------------
__device__ __forceinline__ void copy16_to_lds(const float* __restrict__ g, float* l) {
#if __has_builtin(__builtin_amdgcn_global_load_async_to_lds_b128)
    __builtin_amdgcn_global_load_async_to_lds_b128(
        (g_b128_p)g, (l_b128_p)l, /*offset=*/0, /*cpol=*/0);
#else
    *(float4*)l = *(const float4*)g;   // fallback: sync load + ds_store
#endif
}

__device__ __forceinline__ void wait_async_copies() {
#if __has_builtin(__builtin_amdgcn_s_wait_asynccnt)
    __builtin_amdgcn_s_wait_asynccnt(0);
#elif defined(__gfx1250__)
    asm volatile("s_wait_asynccnt 0" ::: "memory");
#endif
}

// ---------------- degree / dinv ----------------
__global__ void deg_kernel(const int* __restrict__ dst, float* __restrict__ deg, int E) {
    int e = blockIdx.x * blockDim.x + threadIdx.x;
    if (e < E) atomicAdd(&deg[dst[e]], 1.0f);
}

__global__ void dinv_kernel(float* __restrict__ deg, int N) {
    int i = blockIdx.x * blockDim.x + threadIdx.x;
    if (i < N) deg[i] = rsqrtf(deg[i] + 1.0f);   // +1 self loop; always >= 1
}

// ---------------- fp32 WMMA GEMM: C[M,N] = A[M,K] @ W[K,N] ----------------
// One block (N/16 waves) per 16-row strip of A; strip staged in LDS via async
// global->LDS copy; each wave computes one 16x16 tile with V_WMMA_F32_16X16X4_F32.
template<int K, int N>
__global__ void __launch_bounds__(256)
gemm_wmma_kernel(const float* __restrict__ A, const float* __restrict__ W,
                 float* __restrict__ C) {
    const int LDSK = K + 4;                     // padded row stride (bank spread)
    __shared__ float As[16 * LDSK];

    const int mtile = blockIdx.x;

    // Cooperative async copy of the 16xK A-strip into LDS (float4 granules).
    const int NV = (16 * K) / 4;
    for (int i = threadIdx.x; i < NV; i += blockDim.x) {
        int row = i / (K / 4);
        int c4  = (i - row * (K / 4)) * 4;
        copy16_to_lds(A + (size_t)(mtile * 16 + row) * K + c4,
                      As + row * LDSK + c4);
    }
    wait_async_copies();    // per-wave ASYNCcnt
    __syncthreads();        // cross-wave visibility of all strip pieces

    const int ntile = threadIdx.x >> 5;         // wave id = output N tile
    const int lane  = threadIdx.x & 31;
    const int half  = lane >> 4;                // 0: K pair {k0,k0+1}, 1: {k0+2,k0+3}
    const int l     = lane & 15;

    const float* arow = As + l * LDSK + half * 2;            // LDS, b64 reads
    const float* bcol = W + (size_t)(half * 2) * N + ntile * 16 + l;

    v8f acc = {};
    #pragma unroll
    for (int k0 = 0; k0 < K; k0 += 4) {
        // A 16x4 fp32 layout: lanes 0-15 = rows, VGPR0/1 = K {k0,k0+1} (half 0)
        // or {k0+2,k0+3} (half 1)
        v2f a = *(const v2f*)(arow + k0);
        // B 4x16 layout mirrored: lane = column, K-row pair per half-wave
        v2f b;
        b.x = bcol[(size_t)k0 * N];
        b.y = bcol[(size_t)k0 * N + N];
        acc = __builtin_amdgcn_wmma_f32_16x16x4_f32(
                  false, a, false, b, (short)0, acc, false, false);
    }

    // D layout: VGPR i -> row = mtile*16 + half*8 + i, col = ntile*16 + l
    float* crow = C + (size_t)(mtile * 16 + half * 8) * N + ntile * 16 + l;
    #pragma unroll
    for (int i = 0; i < 8; ++i) crow[(size_t)i * N] = acc[i];
}

// ---------------- row scale: h[r,:] *= dinv[r] ----------------
template<int F>
__global__ void scale_rows_kernel(float* __restrict__ h, const float* __restrict__ dinv, int N) {
    const int C = F / 4;
    int t = blockIdx.x * blockDim.x + threadIdx.x;
    if (t >= N * C) return;
    int row = t / C;
    float s = dinv[row];
    float4* p = (float4*)h + t;
    float4 v = *p;
    v.x *= s; v.y *= s; v.z *= s; v.w *= s;
    *p = v;
}

// ---------------- edge scatter: acc[dst,:] += g[src,:] ----------------
template<int F>
__global__ void __launch_bounds__(256)
scatter_kernel(const float* __restrict__ g, const int* __restrict__ src,
               const int* __restrict__ dst, float* __restrict__ acc, int E) {
    const int C = F / 4;                     // float4 chunks per edge
    int t = blockIdx.x * blockDim.x + threadIdx.x;
    if (t >= E * C) return;
    int e = t / C;
    int f = (t - e * C) * 4;
    int s = src[e], d = dst[e];
    float4 v = *(const float4*)(g + (size_t)s * F + f);
    float* o = acc + (size_t)d * F + f;
    atomicAdd(o + 0, v.x);
    atomicAdd(o + 1, v.y);
    atomicAdd(o + 2, v.z);
    atomicAdd(o + 3, v.w);
}

// -------- finalize: y = relu(dinv[r]*(acc + g_selfloop) + b) (in place on acc) --------
template<int F>
__global__ void finalize_kernel(float* __restrict__ acc, const float* __restrict__ g,
                                const float* __restrict__ dinv, const float* __restrict__ bias,
                                int N) {
    const int C = F / 4;
    int t = blockIdx.x * blockDim.x + threadIdx.x;
    if (t >= N * C) return;
    int row = t / C;
    int f = (t - row * C) * 4;
    float s = dinv[row];
    float4 a = ((const float4*)acc)[t];
    float4 gg = ((const float4*)g)[t];
    float4 b = *(const float4*)(bias + f);
    float4 o;
    o.x = fmaxf(fmaf(s, a.x + gg.x, b.x), 0.0f);
    o.y = fmaxf(fmaf(s, a.y + gg.y, b.y), 0.0f);
    o.z = fmaxf(fmaf(s, a.z + gg.z, b.z), 0.0f);
    o.w = fmaxf(fmaf(s, a.w + gg.w, b.w), 0.0f);
    ((float4*)acc)[t] = o;
}

// ---------------- FC (64x16) + log_softmax ----------------
__global__ void __launch_bounds__(256)
fc_lsm_kernel(const float* __restrict__ y, const float* __restrict__ Wfc,
              const float* __restrict__ bfc, float* __restrict__ out, int N) {
    __shared__ float w[HID2 * N_CLS];
    __shared__ float b[N_CLS];
    for (int i = threadIdx.x; i < HID2 * N_CLS; i += blockDim.x) w[i] = Wfc[i];
    if (threadIdx.x < N_CLS) b[threadIdx.x] = bfc[threadIdx.x];
    __syncthreads();

    int n = blockIdx.x * blockDim.x + threadIdx.x;
    if (n >= N) return;
    const float* yr = y + (size_t)n * HID2;

    float logit[N_CLS];
    #pragma unroll
    for (int c = 0; c < N_CLS; ++c) logit[c] = b[c];
    #pragma unroll 4
    for (int k = 0; k < HID2; ++k) {
        float v = yr[k];
        #pragma unroll
        for (int c = 0; c < N_CLS; ++c) logit[c] = fmaf(v, w[k * N_CLS + c], logit[c]);
    }

    float m = logit[0];
    #pragma unroll
    for (int c = 1; c < N_CLS; ++c) m = fmaxf(m, logit[c]);
    float sum = 0.0f;
    #pragma unroll
    for (int c = 0; c < N_CLS; ++c) sum += __expf(logit[c] - m);
    float lse = m + __logf(sum);
    float* orow = out + (size_t)n * N_CLS;
    #pragma unroll
    for (int c = 0; c < N_CLS; ++c) orow[c] = logit[c] - lse;
}

// ---------------- launch ----------------
extern "C" void kernel_launch(void* const* d_in, const int* in_sizes, int n_in,
                              void* d_out, int out_size, void* d_ws, size_t ws_size,
                              hipStream_t stream) {
    const float* x   = (const float*)d_in[0];
    const int*   ei  = (const int*)  d_in[1];   // [2, NE] flat
    const float* W1  = (const float*)d_in[2];
    const float* b1  = (const float*)d_in[3];
    const float* W2  = (const float*)d_in[4];
    const float* b2  = (const float*)d_in[5];
    const float* Wfc = (const float*)d_in[6];
    const float* bfc = (const float*)d_in[7];
    float* out = (float*)d_out;

    const int* src = ei;
    const int* dst = ei + NE;

    // workspace layout (~103 MB): dinv | bufA (N*128) | bufB (N*128)
    float* dinv = (float*)d_ws;
    float* h1   = dinv + ((NN + 255) & ~255);
    float* acc1 = h1 + (size_t)NN * HID;        // y1 after finalize
    float* h2   = h1;                           // reuse: first N*64 of bufA
    float* acc2 = h1 + (size_t)NN * HID2;       // second N*64 of bufA

    const int TPB = 256;
    const int MT  = NN / 16;                    // 6250 strips (exact)

    // degrees -> dinv
    (void)hipMemsetAsync(dinv, 0, (size_t)NN * sizeof(float), stream);
    deg_kernel<<<cdiv(NE, TPB), TPB, 0, stream>>>(dst, dinv, NE);
    dinv_kernel<<<cdiv(NN, TPB), TPB, 0, stream>>>(dinv, NN);

    // ---- layer 1: h1 = x @ W1 ; aggregate ; +b1 ; relu ----
    gemm_wmma_kernel<F_IN, HID><<<MT, (HID / 16) * 32, 0, stream>>>(x, W1, h1);
    scale_rows_kernel<HID><<<cdiv((long long)NN * (HID / 4), TPB), TPB, 0, stream>>>(h1, dinv, NN);
    (void)hipMemsetAsync(acc1, 0, (size_t)NN * HID * sizeof(float), stream);
    scatter_kernel<HID><<<cdiv((long long)NE * (HID / 4), TPB), TPB, 0, stream>>>(h1, src, dst, acc1, NE);
    finalize_kernel<HID><<<cdiv((long long)NN * (HID / 4), TPB), TPB, 0, stream>>>(acc1, h1, dinv, b1, NN);

    // ---- layer 2: h2 = y1 @ W2 ; aggregate ; +b2 ; relu ----
    gemm_wmma_kernel<HID, HID2><<<MT, (HID2 / 16) * 32, 0, stream>>>(acc1, W2, h2);
    scale_rows_kernel<HID2><<<cdiv((long long)NN * (HID2 / 4), TPB), TPB, 0, stream>>>(h2, dinv, NN);
    (void)hipMemsetAsync(acc2, 0, (size_t)NN * HID2 * sizeof(float), stream);
    scatter_kernel<HID2><<<cdiv((long long)NE * (HID2 / 4), TPB), TPB, 0, stream>>>(h2, src, dst, acc2, NE);
    finalize_kernel<HID2><<<cdiv((long long)NN * (HID2 / 4), TPB), TPB, 0, stream>>>(acc2, h2, dinv, b2, NN);

    // ---- FC + log_softmax ----
    fc_lsm_kernel<<<cdiv(NN, TPB), TPB, 0, stream>>>(acc2, Wfc, bfc, out, NN);
}